// BasicBlock_26225070310079
// MI455X (gfx1250) — compile-verified
//
#include <hip/hip_runtime.h>

// ---------------------------------------------------------------------------
// BasicBlock (MinkowskiNet-style): sparse conv -> BN -> ReLU -> sparse conv
// -> BN -> +residual -> ReLU.  f32 end-to-end, WMMA f32_16x16x4 for the GEMMs.
//
// Conv = gather-GEMM-scatter. W[k] (64x64, 16KB) is staged once per block in
// LDS and reused across SLABS_PER_WAVE * 8 waves * 16 rows = 512 rows, cutting
// weight re-staging L2 traffic 4x vs one-slab-per-wave. Inner product uses
// V_WMMA_F32_16X16X4_F32 (f32 keeps reference precision; the op is L2
// gather/scatter bound, not matrix-rate bound, so f32 WMMA is the right
// precision/throughput point).
// ---------------------------------------------------------------------------

typedef float v2f __attribute__((ext_vector_type(2)));
typedef float v8f __attribute__((ext_vector_type(8)));

#define CCH 64          // channels
#define LDW 68          // LDS row stride (floats) to avoid bank conflicts
#define ROWS_PER_WAVE 16
#define WAVES_PER_BLOCK 8
#define SLABS_PER_WAVE 4
#define ROWS_PER_BLOCK (ROWS_PER_WAVE * WAVES_PER_BLOCK)            // 128
#define ROWS_PER_CTA   (ROWS_PER_BLOCK * SLABS_PER_WAVE)            // 512

// ---------------------------------------------------------------------------
// Zero the accumulators + BN stat buffers (float4 granularity).
// ---------------------------------------------------------------------------
__global__ void zero_kernel(float4* __restrict__ p, long n4) {
    long i = (long)blockIdx.x * blockDim.x + threadIdx.x;
    if (i < n4) p[i] = float4{0.f, 0.f, 0.f, 0.f};
}

// ---------------------------------------------------------------------------
// Gather-GEMM-scatter sparse conv.
//   Y[out_maps[k,m], :] += X[in_maps[k,m], :] @ W[k]       (atomic scatter)
// Each wave: 4 slabs of 16 rows; per slab 4 col-tiles x 16 K-steps = 64 WMMAs.
// ---------------------------------------------------------------------------
__global__ void __launch_bounds__(256)
spconv_wmma_kernel(const float* __restrict__ X,
                   const float* __restrict__ W,
                   const int*  __restrict__ imap,
                   const int*  __restrict__ omap,
                   float* __restrict__ Y,
                   int M) {
    __shared__ float sw[64 * LDW];

    const int k = blockIdx.y;
    const float* Wk = W + (size_t)k * CCH * CCH;

    // Cooperative load of W[k] (64x64 f32 = 16 KB) into LDS, float4 vectors.
    for (int i = threadIdx.x; i < 1024; i += 256) {
        int r  = i >> 4;          // row 0..63
        int c4 = (i & 15) << 2;   // col 0..60 step 4
        float4 v = ((const float4*)Wk)[i];
        *(float4*)&sw[r * LDW + c4] = v;
    }
    __syncthreads();

    const int wave = threadIdx.x >> 5;   // wave32
    const int lane = threadIdx.x & 31;
    const int lr = lane & 15;            // row within 16-row tile
    const int hi = lane >> 4;            // K-phase select (+2 for upper half)
    const int blockBase = blockIdx.x * ROWS_PER_CTA;
    const float* bcol = &sw[(hi * 2) * LDW + lr];

    for (int s = 0; s < SLABS_PER_WAVE; ++s) {
        const int m0 = blockBase + s * ROWS_PER_BLOCK + wave * ROWS_PER_WAVE;
        if (m0 >= M) continue;           // M % 16 == 0 -> whole slabs only

        // --- Gather A fragments: lane holds cols {4t+2hi, 4t+2hi+1} of row lr.
        const int gidx = imap[k * M + m0 + lr];
        const float* src = X + (size_t)gidx * CCH + hi * 2;
        v2f a[16];
#pragma unroll
        for (int t = 0; t < 16; ++t)
            a[t] = *(const v2f*)(src + 4 * t);  // 8B load; lr/lr+16 adjacent

        // --- Output row indices this lane scatters to (rows r + 8*hi).
        int oidx[8];
#pragma unroll
        for (int r = 0; r < 8; ++r)
            oidx[r] = omap[k * M + m0 + r + 8 * hi];

        // --- 4 column tiles of 16; 16 WMMA K-steps each.
#pragma unroll
        for (int j = 0; j < 4; ++j) {
            v8f acc = {};
            const float* bj = bcol + j * 16;
#pragma unroll
            for (int t = 0; t < 16; ++t) {
                v2f b;
                b.x = bj[(4 * t)     * LDW];   // W[4t + 2hi    ][16j + lr]
                b.y = bj[(4 * t + 1) * LDW];   // W[4t + 2hi + 1][16j + lr]
                acc = __builtin_amdgcn_wmma_f32_16x16x4_f32(
                    false, a[t], false, b, (short)0, acc, false, false);
            }
            // Scatter-add: VGPR r holds D[row = r + 8*hi][col = 16j + lr].
#pragma unroll
            for (int r = 0; r < 8; ++r)
                atomicAdd(&Y[(size_t)oidx[r] * CCH + j * 16 + lr], acc[r]);
        }
    }
}

// ---------------------------------------------------------------------------
// Per-channel sum / sum-of-squares over the [N,64] matrix.
// stats[0:64] = sum, stats[64:128] = sumsq  (pre-zeroed).
// ---------------------------------------------------------------------------
__global__ void __launch_bounds__(256)
bn_stats_kernel(const float* __restrict__ x, float* __restrict__ stats, int n) {
    __shared__ float s_sum[256];
    __shared__ float s_sq[256];
    const int c  = threadIdx.x & 63;
    const int rt = threadIdx.x >> 6;                 // 0..3
    const int stride = (256 >> 6) * gridDim.x;       // 4 * gridDim
    float sum = 0.f, sq = 0.f;
    for (int r = blockIdx.x * 4 + rt; r < n; r += stride) {
        float v = x[(size_t)r * CCH + c];
        sum += v;
        sq  += v * v;
    }
    s_sum[threadIdx.x] = sum;
    s_sq [threadIdx.x] = sq;
    __syncthreads();
    if (threadIdx.x < 64) {
        float ts = s_sum[c] + s_sum[c + 64] + s_sum[c + 128] + s_sum[c + 192];
        float tq = s_sq [c] + s_sq [c + 64] + s_sq [c + 128] + s_sq [c + 192];
        atomicAdd(&stats[c],      ts);
        atomicAdd(&stats[64 + c], tq);
    }
}

// ---------------------------------------------------------------------------
// In-place BN (batch stats) + ReLU.
// ---------------------------------------------------------------------------
__global__ void __launch_bounds__(256)
bn_relu_kernel(float* __restrict__ x, const float* __restrict__ stats,
               const float* __restrict__ gamma, const float* __restrict__ beta,
               int n) {
    int i = blockIdx.x * blockDim.x + threadIdx.x;
    if (i >= n * CCH) return;
    int c = i & 63;
    float invn = 1.0f / (float)n;
    float mean = stats[c] * invn;
    float var  = stats[64 + c] * invn - mean * mean;
    float inv  = rsqrtf(var + 1e-5f);
    float v = (x[i] - mean) * inv * gamma[c] + beta[c];
    x[i] = v > 0.f ? v : 0.f;
}

// ---------------------------------------------------------------------------
// Final: BN2 + residual + ReLU -> d_out.
// ---------------------------------------------------------------------------
__global__ void __launch_bounds__(256)
bn_add_relu_kernel(const float* __restrict__ x, const float* __restrict__ resid,
                   const float* __restrict__ stats,
                   const float* __restrict__ gamma, const float* __restrict__ beta,
                   float* __restrict__ out, int n) {
    int i = blockIdx.x * blockDim.x + threadIdx.x;
    if (i >= n * CCH) return;
    int c = i & 63;
    float invn = 1.0f / (float)n;
    float mean = stats[c] * invn;
    float var  = stats[64 + c] * invn - mean * mean;
    float inv  = rsqrtf(var + 1e-5f);
    float v = (x[i] - mean) * inv * gamma[c] + beta[c] + resid[i];
    out[i] = v > 0.f ? v : 0.f;
}

// ---------------------------------------------------------------------------
// Launch: zero -> conv1 -> stats1 -> bn1+relu -> conv2 -> stats2 -> bn2+res+relu
// ---------------------------------------------------------------------------
extern "C" void kernel_launch(void* const* d_in, const int* in_sizes, int n_in,
                              void* d_out, int out_size, void* d_ws, size_t ws_size,
                              hipStream_t stream) {
    const float* feat = (const float*)d_in[0];
    const float* W1   = (const float*)d_in[1];
    const float* g1   = (const float*)d_in[2];
    const float* b1   = (const float*)d_in[3];
    const float* W2   = (const float*)d_in[4];
    const float* g2   = (const float*)d_in[5];
    const float* b2   = (const float*)d_in[6];
    const int*   imap = (const int*)d_in[7];
    const int*   omap = (const int*)d_in[8];

    const int N = in_sizes[0] / CCH;                 // 100000
    const int K = in_sizes[1] / (CCH * CCH);         // 27
    const int M = in_sizes[7] / K;                   // 50000

    // Workspace layout: out1 [N*64] | out2 [N*64] | stats1 [128] | stats2 [128]
    float* out1   = (float*)d_ws;
    float* out2   = out1 + (size_t)N * CCH;
    float* stats1 = out2 + (size_t)N * CCH;
    float* stats2 = stats1 + 128;

    // Zero accumulators + stats.
    long n4 = ((long)2 * N * CCH + 256) / 4;
    zero_kernel<<<(unsigned)((n4 + 255) / 256), 256, 0, stream>>>((float4*)d_ws, n4);

    dim3 cgrid((M + ROWS_PER_CTA - 1) / ROWS_PER_CTA, K);
    int totalNC = N * CCH;
    int eblocks = (totalNC + 255) / 256;

    // conv1 + BN1 + ReLU (in place on out1)
    spconv_wmma_kernel<<<cgrid, 256, 0, stream>>>(feat, W1, imap, omap, out1, M);
    bn_stats_kernel<<<512, 256, 0, stream>>>(out1, stats1, N);
    bn_relu_kernel<<<eblocks, 256, 0, stream>>>(out1, stats1, g1, b1, N);

    // conv2 + BN2 + residual + ReLU -> d_out
    spconv_wmma_kernel<<<cgrid, 256, 0, stream>>>(out1, W2, imap, omap, out2, M);
    bn_stats_kernel<<<512, 256, 0, stream>>>(out2, stats2, N);
    bn_add_relu_kernel<<<eblocks, 256, 0, stream>>>(out2, feat, stats2, g2, b2,
                                                    (float*)d_out, N);
}